// GCN_29411936043363
// MI455X (gfx1250) — compile-verified
//
#include <hip/hip_runtime.h>
#include <hip/hip_bf16.h>
#include <cstddef>
#include <cmath>

typedef float v2f __attribute__((ext_vector_type(2)));
typedef float v8f __attribute__((ext_vector_type(8)));

#define HH 128   // hidden dim (== in dim)

// ---------------- degree + norm ----------------
__global__ void deg_kernel(const int* __restrict__ src, const int* __restrict__ dst,
                           float* __restrict__ deg_out, float* __restrict__ deg_in, int E) {
    int e = blockIdx.x * blockDim.x + threadIdx.x;
    if (e < E) {
        atomicAdd(&deg_out[src[e]], 1.0f);
        atomicAdd(&deg_in[dst[e]], 1.0f);
    }
}

__global__ void norm_kernel(float* __restrict__ a, float* __restrict__ b, int n) {
    int i = blockIdx.x * blockDim.x + threadIdx.x;
    if (i < n) {
        float x = a[i]; a[i] = (x > 0.f) ? 1.0f / sqrtf(x) : 1.0f;
        float y = b[i]; b[i] = (y > 0.f) ? 1.0f / sqrtf(y) : 1.0f;
    }
}

// ---------------- edge gather + scatter-add ----------------
// one wave32 per edge: coalesced float4 row read of h[src] (scaled by norm_src),
// 4 f32 atomic adds per lane into agg[dst]  (both rows L2-resident: 51.2MB < 192MB L2)
__global__ void gather_scatter(const float* __restrict__ h, const int* __restrict__ src,
                               const int* __restrict__ dst, const float* __restrict__ norm_src,
                               float* __restrict__ agg, int E) {
    int e = blockIdx.x * (blockDim.x >> 5) + (threadIdx.x >> 5);
    if (e >= E) return;
    int lane = threadIdx.x & 31;
    int s = src[e], d = dst[e];
    float ns = norm_src[s];
    const float4* hp = (const float4*)(h + (size_t)s * HH);
    float4 v = hp[lane];
    float* op = agg + (size_t)d * HH + (size_t)lane * 4;
    atomicAdd(op + 0, v.x * ns);
    atomicAdd(op + 1, v.y * ns);
    atomicAdd(op + 2, v.z * ns);
    atomicAdd(op + 3, v.w * ns);
}

// ---------------- WMMA f32 GEMM: out[r][c] = act( (in[r]*scale[r]) @ W + bias[c] ) ----------------
// in: nrows x 128 row-major. W: 128 x (KT*16) row-major. out: nrows x (KT*16).
// One wave per 16-row M-tile; K=128 consumed as 32 x v_wmma_f32_16x16x4_f32.
// In-place safe (out==in): tile fully staged to LDS before any store.
template<int KT, bool ELU>
__global__ __launch_bounds__(128) void wmma_gemm(const float* __restrict__ in,
                                                 const float* __restrict__ scale,
                                                 const float* __restrict__ W,
                                                 const float* __restrict__ bias,
                                                 float* __restrict__ out, int nrows) {
    __shared__ float sA[4][16 * HH];          // 32 KB: one 16x128 tile per wave
    const int wave = threadIdx.x >> 5;
    const int lane = threadIdx.x & 31;
    const int row0 = (blockIdx.x * 4 + wave) * 16;
    const bool active = (row0 < nrows);       // wave-uniform

    float* tile = sA[wave];
    if (active) {
        // stage 16x128 tile, fused norm_dst scaling; coalesced 128B/wave reads
        for (int i = 0; i < 64; ++i) {
            int idx = i * 32 + lane;
            int r = idx >> 7, c = idx & (HH - 1);
            float s = scale ? scale[row0 + r] : 1.0f;
            tile[idx] = in[(size_t)(row0 + r) * HH + c] * s;
        }
    }
    __syncthreads();                          // LDS visibility across lanes
    if (!active) return;

    v8f acc[KT];
    #pragma unroll
    for (int t = 0; t < KT; ++t) acc[t] = v8f{0.f, 0.f, 0.f, 0.f, 0.f, 0.f, 0.f, 0.f};

    const int half = lane >> 4;               // 0: K={0,1}, 1: K={2,3}
    const int lr   = lane & 15;               // A: row M, B/C: col N
    const int wstride = KT * 16;

    for (int kc = 0; kc < 32; ++kc) {
        const int k0 = kc * 4;
        // A fragment (16x4 f32): lane half selects K pair, lr selects row
        const float* ap = &tile[lr * HH + k0 + half * 2];
        v2f a; a.x = ap[0]; a.y = ap[1];
        const float* bp = W + (size_t)(k0 + half * 2) * wstride + lr;
        #pragma unroll
        for (int t = 0; t < KT; ++t) {
            // B fragment (4x16 f32): row K striped across lane halves, col = t*16+lr
            v2f b; b.x = bp[t * 16]; b.y = bp[t * 16 + wstride];
            acc[t] = __builtin_amdgcn_wmma_f32_16x16x4_f32(
                false, a, false, b, (short)0, acc[t], false, false);
        }
    }

    // C/D layout: VGPR v -> row (v + half*8), col = t*16 + lr
    #pragma unroll
    for (int t = 0; t < KT; ++t) {
        int c = t * 16 + lr;
        float bv = bias[c];
        float* op = out + (size_t)(row0 + half * 8) * wstride + c;
        #pragma unroll
        for (int v = 0; v < 8; ++v) {
            float x = acc[t][v] + bv;
            if (ELU) x = (x > 0.f) ? x : (expf(x) - 1.0f);
            op[(size_t)v * wstride] = x;
        }
    }
}

// ---------------- launch ----------------
extern "C" void kernel_launch(void* const* d_in, const int* in_sizes, int n_in,
                              void* d_out, int out_size, void* d_ws, size_t ws_size,
                              hipStream_t stream) {
    const float* x      = (const float*)d_in[0];
    const int*   src    = (const int*)d_in[1];
    const int*   dst    = (const int*)d_in[2];
    const float* conv_w = (const float*)d_in[3];   // (3,128,128)
    const float* conv_b = (const float*)d_in[4];   // (3,128)
    const float* lin_w  = (const float*)d_in[5];   // (128,64)
    const float* lin_b  = (const float*)d_in[6];   // (64,)
    float* out = (float*)d_out;

    const int N = in_sizes[0] / HH;
    const int E = in_sizes[1];

    float* ws = (float*)d_ws;
    float* norm_src = ws;                          // N
    float* norm_dst = ws + N;                      // N
    float* buf0 = ws + 2 * (size_t)N;              // N*128
    float* buf1 = buf0 + (size_t)N * HH;           // N*128
    const size_t bufBytes = (size_t)N * HH * sizeof(float);

    // degrees -> norms
    hipMemsetAsync(norm_src, 0, 2 * (size_t)N * sizeof(float), stream);
    deg_kernel<<<(E + 255) / 256, 256, 0, stream>>>(src, dst, norm_src, norm_dst, E);
    norm_kernel<<<(N + 255) / 256, 256, 0, stream>>>(norm_src, norm_dst, N);

    const int ggrid = (E + 7) / 8;                 // 8 edges (waves) per 256-thread block
    const int mgrid = (N + 63) / 64;               // 4 waves x 16 rows per block

    // layer 0: x -> buf0
    hipMemsetAsync(buf0, 0, bufBytes, stream);
    gather_scatter<<<ggrid, 256, 0, stream>>>(x, src, dst, norm_src, buf0, E);
    wmma_gemm<8, true><<<mgrid, 128, 0, stream>>>(buf0, norm_dst, conv_w + 0 * HH * HH,
                                                  conv_b + 0 * HH, buf0, N);
    // layer 1: buf0 -> buf1
    hipMemsetAsync(buf1, 0, bufBytes, stream);
    gather_scatter<<<ggrid, 256, 0, stream>>>(buf0, src, dst, norm_src, buf1, E);
    wmma_gemm<8, true><<<mgrid, 128, 0, stream>>>(buf1, norm_dst, conv_w + 1 * HH * HH,
                                                  conv_b + 1 * HH, buf1, N);
    // layer 2: buf1 -> buf0 (ELU applied here since final linear consumes elu(h3))
    hipMemsetAsync(buf0, 0, bufBytes, stream);
    gather_scatter<<<ggrid, 256, 0, stream>>>(buf1, src, dst, norm_src, buf0, E);
    wmma_gemm<8, true><<<mgrid, 128, 0, stream>>>(buf0, norm_dst, conv_w + 2 * HH * HH,
                                                  conv_b + 2 * HH, buf0, N);
    // final linear: elu(h3) @ lin_w + lin_b -> out (N x 64)
    wmma_gemm<4, false><<<mgrid, 128, 0, stream>>>(buf0, nullptr, lin_w, lin_b, out, N);
}